// Outter2dFusion_2413771620362
// MI455X (gfx1250) — compile-verified
//
#include <hip/hip_runtime.h>
#include <hip/hip_bf16.h>

// Outer-product fusion: out[row, i, j] = gate[row, i] * features[row, j]
// rows = B*S = 2048, D = 256. Output = 512 MiB f32 -> store-bandwidth bound.
// Compute each 16x16 output tile with V_WMMA_F32_16X16X4_F32 (rank-1 GEMM):
//   A(16x4) = gate segment replicated across all K slots
//   B(4x16) = features segment in one K row, zeros elsewhere
//   D = A x B = exact f32 outer product (single RNE multiply per element).

typedef __attribute__((ext_vector_type(2))) float v2f;
typedef __attribute__((ext_vector_type(8))) float v8f;

#define DDIM 256

__global__ __launch_bounds__(512) void outer_wmma_kernel(
    const float* __restrict__ gate,
    const float* __restrict__ feat,
    float* __restrict__ out) {
  const int row  = blockIdx.x;          // 0..2047 (flattened B*S)
  const int lane = threadIdx.x & 31;    // wave32
  const int wave = threadIdx.x >> 5;    // 0..15 -> i-block index
  const int i0   = wave << 4;           // 16-row block of output rows
  const int n    = lane & 15;           // N index within tile
  const int half = lane >> 4;           // 0: lanes 0-15, 1: lanes 16-31

  // ---- A matrix: 16x4 f32, 2 VGPRs/lane.
  // Layout (ISA 7.12.2): lanes 0-15 hold K=0 (v0), K=1 (v1) for M=lane;
  // lanes 16-31 hold K=2 (v0), K=3 (v1) for M=lane-16.
  // Replicate gate[m] across all K slots -> robust: D[m][n] = sum_k A[m][k]*B[k][n]
  // picks up gate[m]*feat[n] from whichever single K-row B is non-zero in.
  const float g = gate[row * DDIM + i0 + n];
  v2f a;
  a.x = g;
  a.y = g;

  const float* frow = feat + row * DDIM;
  // D layout: VGPR v holds (M = v + 8*half, N = n). Per-lane store base:
  float* obase = out + (size_t)row * (DDIM * DDIM)
                     + (size_t)(i0 + half * 8) * DDIM + n;

  for (int jb = 0; jb < 16; ++jb) {
    const int j0 = jb << 4;

    // ---- B matrix: 4x16 f32, 2 VGPRs/lane. Non-zero only in one K row:
    // VGPR0, lanes 0-15 (a complete row K, N = lane). Everything else 0.
    const float f = frow[j0 + n];
    v2f b;
    b.x = (half == 0) ? f : 0.0f;
    b.y = 0.0f;

    v8f c = {};
    // 8 args: (neg_a, A, neg_b, B, c_mod, C, reuse_a, reuse_b)
    v8f d = __builtin_amdgcn_wmma_f32_16x16x4_f32(
        false, a, false, b, (short)0, c, false, false);

    float* op = obase + j0;
#pragma unroll
    for (int v = 0; v < 8; ++v) {
      op[v * DDIM] = d[v];   // global_store_b32 with immediate offsets v*1024B
    }
  }
}

extern "C" void kernel_launch(void* const* d_in, const int* in_sizes, int n_in,
                              void* d_out, int out_size, void* d_ws, size_t ws_size,
                              hipStream_t stream) {
  const float* gate = (const float*)d_in[0];
  const float* feat = (const float*)d_in[1];
  float* out = (float*)d_out;

  const int rows = in_sizes[0] / DDIM;  // B*S = 2048

  dim3 grid(rows);
  dim3 block(512);                      // 16 waves -> 16 i-blocks per row
  outer_wmma_kernel<<<grid, block, 0, stream>>>(gate, feat, out);
}